// EIGNN_finite_52733608460981
// MI455X (gfx1250) — compile-verified
//
#include <hip/hip_runtime.h>
#include <math.h>

typedef __attribute__((ext_vector_type(2))) float v2f;
typedef __attribute__((ext_vector_type(8))) float v8f;

#define MDIM   128
#define GAMMA  0.8f
#define EPS_F  1e-6f
#define BN_EPS 1e-5f
#define K_ITERS 5
#define MY     10

// ---------------- Y0 = X^T  ([128,N] -> [N,128]) ----------------
__global__ void k_transpose(const float* __restrict__ X, float* __restrict__ Y0, int N) {
  __shared__ float tile[32][33];
  int n0 = blockIdx.x * 32;
  int m0 = blockIdx.y * 32;
  int tx = threadIdx.x, ty = threadIdx.y;        // block (32,8)
  #pragma unroll
  for (int j = 0; j < 32; j += 8)
    tile[ty + j][tx] = X[(size_t)(m0 + ty + j) * N + n0 + tx];
  __syncthreads();
  #pragma unroll
  for (int j = 0; j < 32; j += 8)
    Y0[(size_t)(n0 + ty + j) * MDIM + m0 + tx] = tile[tx][ty + j];
}

// ---------------- FF = F^T F  (128x128, symmetric) ----------------
__global__ void k_ff(const float* __restrict__ F, float* __restrict__ FF) {
  __shared__ float fcol[128];
  int i = blockIdx.x, j = threadIdx.x;           // 128 blocks x 128 threads
  fcol[j] = F[(size_t)j * MDIM + i];             // column i of F
  __syncthreads();
  float s = 0.f;
  #pragma unroll 8
  for (int k = 0; k < MDIM; ++k)
    s = fmaf(fcol[k], F[(size_t)k * MDIM + j], s);
  FF[i * MDIM + j] = s;
}

// ---------------- G = FF / (||FF||_F + eps) ----------------
__global__ void k_gnorm(const float* __restrict__ FF, float* __restrict__ G) {
  __shared__ double red[256];
  __shared__ float invn;
  int t = threadIdx.x;                            // 1 block x 256 threads
  double s = 0.0;
  for (int i = t; i < MDIM * MDIM; i += 256) { double v = FF[i]; s += v * v; }
  red[t] = s; __syncthreads();
  for (int w = 128; w > 0; w >>= 1) { if (t < w) red[t] += red[t + w]; __syncthreads(); }
  if (t == 0) invn = 1.0f / ((float)sqrt(red[0]) + EPS_F);
  __syncthreads();
  float inv = invn;
  for (int i = t; i < MDIM * MDIM; i += 256) G[i] = FF[i] * inv;
}

// ---------------- SpMM: SY[r,:] += val * Y[c,:]  (wave per edge) ----------------
__global__ void k_spmm(const float* __restrict__ Ysrc, const float* __restrict__ val,
                       const int* __restrict__ row, const int* __restrict__ col,
                       float* __restrict__ SY, int E) {
  int lane  = threadIdx.x & 31;
  int wave  = (blockIdx.x * blockDim.x + threadIdx.x) >> 5;
  int nwave = (gridDim.x * blockDim.x) >> 5;
  for (int e = wave; e < E; e += nwave) {
    int c = col[e];
    int r = row[e];
    float v = val[e];
    int en = e + nwave;
    if (en < E)  // hide L2 gather latency: global_prefetch_b8 on next source row
      __builtin_prefetch(Ysrc + (size_t)col[en] * MDIM + lane * 4, 0, 0);
    const float4* ys = (const float4*)(Ysrc + (size_t)c * MDIM);
    float4 y = ys[lane];                          // 32 lanes x 16B = 512B coalesced
    float* dst = SY + (size_t)r * MDIM + lane * 4;
    atomicAdd(dst + 0, v * y.x);
    atomicAdd(dst + 1, v * y.y);
    atomicAdd(dst + 2, v * y.z);
    atomicAdd(dst + 3, v * y.w);
  }
}

// ---------------- Yout = GAMMA * (SY @ G) + Y0   via V_WMMA_F32_16X16X4_F32 ----------------
// block = 256 threads = 8 waves; block per 16-row n-tile; wave w does cols [16w,16w+16)
__global__ void k_wmma_gemm(const float* __restrict__ SY, const float* __restrict__ G,
                            const float* __restrict__ Y0, float* __restrict__ Yout, int N) {
  int lane = threadIdx.x & 31;
  int wv   = threadIdx.x >> 5;                    // 0..7
  int n0   = blockIdx.x * 16;
  int f0   = wv * 16;
  int m    = lane & 15;                           // A row / B col within tile
  int kk   = (lane >> 4) << 1;                    // 0 (lanes 0-15) or 2 (lanes 16-31)
  v8f c = {0.f, 0.f, 0.f, 0.f, 0.f, 0.f, 0.f, 0.f};
  const float* arow = SY + (size_t)(n0 + m) * MDIM + kk;   // A: 16x4 f32 tiles
  const float* brow = G  + (size_t)(f0 + m) * MDIM + kk;   // B: G row (G symmetric)
  #pragma unroll
  for (int k = 0; k < MDIM; k += 4) {
    v2f a = *(const v2f*)(arow + k);              // K = kk, kk+1
    v2f b = *(const v2f*)(brow + k);
    c = __builtin_amdgcn_wmma_f32_16x16x4_f32(false, a, false, b,
                                              (short)0, c, false, false);
  }
  int half = lane >> 4;
  int colf = lane & 15;
  #pragma unroll
  for (int v = 0; v < 8; ++v) {                   // D layout: VGPR v -> row v + 8*half
    size_t idx = (size_t)(n0 + v + 8 * half) * MDIM + f0 + colf;
    Yout[idx] = GAMMA * c[v] + Y0[idx];
  }
}

// ---------------- BN stats (f64 accumulation) ----------------
__global__ void k_bn_stats(const float* __restrict__ Y, double* __restrict__ stats, int N) {
  int f = threadIdx.x;                            // 128 threads: one feature each
  int rows_per = (N + gridDim.x - 1) / gridDim.x;
  int nbeg = blockIdx.x * rows_per;
  int nend = nbeg + rows_per; if (nend > N) nend = N;
  double s = 0.0, s2 = 0.0;
  for (int n = nbeg; n < nend; ++n) {
    double x = (double)Y[(size_t)n * MDIM + f];   // coalesced 512B per row
    s += x; s2 += x * x;
  }
  atomicAdd(&stats[f], s);
  atomicAdd(&stats[MDIM + f], s2);
}

// ---------------- fold BN into projection: Bs = B*scale, biasY = B@shift ----------------
__global__ void k_bn_fold(const double* __restrict__ stats, const float* __restrict__ w,
                          const float* __restrict__ bb, const float* __restrict__ B,
                          float* __restrict__ Bs, float* __restrict__ biasY, int N) {
  __shared__ float sh[128];
  int f = threadIdx.x;                            // 128 threads
  double mean = stats[f] / (double)N;
  double var  = stats[MDIM + f] / (double)N - mean * mean;
  float sc  = w[f] * rsqrtf((float)var + BN_EPS);
  float shv = bb[f] - (float)mean * sc;
  sh[f] = shv;
  #pragma unroll
  for (int y = 0; y < MY; ++y) Bs[y * MDIM + f] = B[y * MDIM + f] * sc;
  __syncthreads();
  if (f < MY) {
    float acc = 0.f;
    for (int j = 0; j < MDIM; ++j) acc = fmaf(sh[j], B[f * MDIM + j], acc);
    biasY[f] = acc;
  }
}

// ---------------- out[n,:] = biasY + Y[n,:] @ Bs^T ----------------
__global__ void k_final(const float* __restrict__ Y, const float* __restrict__ Bs,
                        const float* __restrict__ biasY, float* __restrict__ out, int N) {
  __shared__ float sB[MY * MDIM];
  __shared__ float sb[MY];
  for (int i = threadIdx.x; i < MY * MDIM; i += blockDim.x) sB[i] = Bs[i];
  if (threadIdx.x < MY) sb[threadIdx.x] = biasY[threadIdx.x];
  __syncthreads();
  for (int n = blockIdx.x * blockDim.x + threadIdx.x; n < N; n += gridDim.x * blockDim.x) {
    float acc[MY];
    #pragma unroll
    for (int y = 0; y < MY; ++y) acc[y] = sb[y];
    const float4* yr = (const float4*)(Y + (size_t)n * MDIM);
    #pragma unroll 4
    for (int q = 0; q < MDIM / 4; ++q) {
      float4 v = yr[q];
      #pragma unroll
      for (int y = 0; y < MY; ++y) {
        const float* bp = sB + y * MDIM + q * 4;
        acc[y] = fmaf(v.x, bp[0], fmaf(v.y, bp[1], fmaf(v.z, bp[2], fmaf(v.w, bp[3], acc[y]))));
      }
    }
    float* op = out + (size_t)n * MY;
    #pragma unroll
    for (int y = 0; y < MY; ++y) op[y] = acc[y];
  }
}

extern "C" void kernel_launch(void* const* d_in, const int* in_sizes, int n_in,
                              void* d_out, int out_size, void* d_ws, size_t ws_size,
                              hipStream_t stream) {
  const float* X    = (const float*)d_in[0];
  const float* F    = (const float*)d_in[1];
  const float* B    = (const float*)d_in[2];
  const float* bnw  = (const float*)d_in[3];
  const float* bnb  = (const float*)d_in[4];
  const float* eval = (const float*)d_in[5];
  const int*   erow = (const int*)d_in[6];
  const int*   ecol = (const int*)d_in[7];
  float* out = (float*)d_out;

  int N = in_sizes[0] / MDIM;     // 100000
  int E = in_sizes[5];            // 1600000

  auto al = [](size_t x) { return (x + 255) & ~(size_t)255; };
  char* ws = (char*)d_ws;
  size_t big = al((size_t)N * MDIM * sizeof(float));
  float*  SY    = (float*)ws;  ws += big;
  float*  Ycur  = (float*)ws;  ws += big;
  float*  Y0    = (float*)ws;  ws += big;
  float*  FFb   = (float*)ws;  ws += al(MDIM * MDIM * sizeof(float));
  float*  Gm    = (float*)ws;  ws += al(MDIM * MDIM * sizeof(float));
  double* stats = (double*)ws; ws += al(2 * MDIM * sizeof(double));
  float*  Bs    = (float*)ws;  ws += al(MY * MDIM * sizeof(float));
  float*  biasY = (float*)ws;  ws += al(MY * sizeof(float));

  // Preprocess: X^T and normalized Gram matrix
  dim3 bT(32, 8), gT(N / 32, MDIM / 32);
  k_transpose<<<gT, bT, 0, stream>>>(X, Y0, N);
  k_ff<<<MDIM, MDIM, 0, stream>>>(F, FFb);
  k_gnorm<<<1, 256, 0, stream>>>(FFb, Gm);

  // K fixed-point iterations
  for (int it = 0; it < K_ITERS; ++it) {
    hipMemsetAsync(SY, 0, (size_t)N * MDIM * sizeof(float), stream);
    const float* src = (it == 0) ? Y0 : Ycur;
    k_spmm<<<2048, 256, 0, stream>>>(src, eval, erow, ecol, SY, E);
    k_wmma_gemm<<<N / 16, 256, 0, stream>>>(SY, Gm, Y0, Ycur, N);
  }

  // BatchNorm (training stats) folded into final projection
  hipMemsetAsync(stats, 0, 2 * MDIM * sizeof(double), stream);
  k_bn_stats<<<256, MDIM, 0, stream>>>(Ycur, stats, N);
  k_bn_fold<<<1, MDIM, 0, stream>>>(stats, bnw, bnb, B, Bs, biasY, N);
  k_final<<<512, 256, 0, stream>>>(Ycur, Bs, biasY, out, N);
}